// SAGEConv_38001870635073
// MI455X (gfx1250) — compile-verified
//
#include <hip/hip_runtime.h>
#include <hip/hip_bf16.h>

typedef __attribute__((ext_vector_type(2))) float v2f;
typedef __attribute__((ext_vector_type(8))) float v8f;

#define N_DST_C 50000
#define D_C 64

// ---------------- Phase 1: zero the accumulators in workspace ----------------
__global__ void sage_zero_ws(float4* p, int n4) {
    int i = blockIdx.x * blockDim.x + threadIdx.x;
    if (i < n4) p[i] = make_float4(0.f, 0.f, 0.f, 0.f);
}

// ---------------- Phase 2: edge scatter (segment-sum + degree) ---------------
// One thread per (edge, feature). A wave32 covers 32 consecutive features of a
// single edge -> fully coalesced 128B gather from feat[src] and coalesced
// atomic adds into nsum[dst].
__global__ void sage_scatter(const float* __restrict__ feat,
                             const int* __restrict__ esrc,
                             const int* __restrict__ edst,
                             float* __restrict__ nsum,
                             float* __restrict__ deg,
                             int E) {
    int idx = blockIdx.x * blockDim.x + threadIdx.x;
    int e = idx >> 6;           // 64 features per edge
    int f = idx & 63;
    if (e >= E) return;
    int s = esrc[e];
    int d = edst[e];
    atomicAdd(&nsum[(size_t)d * D_C + f], feat[(size_t)s * D_C + f]);
    if (f == 0) atomicAdd(&deg[d], 1.0f);
}

// ---------------- Phase 3: fused dual-GEMM + bias via f32 WMMA ---------------
// rst = feat[self_src] @ Wself^T + (nsum/deg) @ Wneigh^T + bself + bneigh
// One wave per 16x16 output tile. K=64 consumed in 16 steps of
// V_WMMA_F32_16X16X4_F32 (x2 for the two weight matrices), accumulator chained.
//
// A 16x4 f32 layout: lanes 0-15 -> M=lane (K=k0+0/1 in a[0]/a[1]);
//                    lanes 16-31 -> M=lane-16 (K=k0+2/3).
// B 4x16 f32 layout (mirrored): lane holds column N=lane&15, same K split.
// Since W is [D_OUT, D_IN] row-major, B[k][n] = W[n*64 + k] -> per-lane
// contiguous float2, same addressing pattern as A.
__global__ void sage_gemm_wmma(const float* __restrict__ feat,
                               const int* __restrict__ self_src,
                               const float* __restrict__ nsum,
                               const float* __restrict__ deg,
                               const float* __restrict__ Wself,
                               const float* __restrict__ bself,
                               const float* __restrict__ Wneigh,
                               const float* __restrict__ bneigh,
                               float* __restrict__ out) {
    int gw   = (blockIdx.x * blockDim.x + threadIdx.x) >> 5;  // global wave id
    int lane = threadIdx.x & 31;
    const int tilesN = D_C / 16;            // 4 column tiles
    int tm = gw / tilesN;                   // 0..3124
    int tn = gw % tilesN;
    if (tm >= N_DST_C / 16) return;         // wave-uniform exit (EXEC stays all-1)

    int half = lane >> 4;                   // 0: K pair {0,1}, 1: K pair {2,3}
    int l15  = lane & 15;
    int row  = tm * 16 + l15;               // dst node index (A-matrix M)
    int col  = tn * 16 + l15;               // output feature (B-matrix N)

    int   srow = self_src[row];             // identity permutation, honored anyway
    float rdeg = 1.0f / fmaxf(deg[row], 1.0f);

    const float* fs = feat + (size_t)srow * D_C;
    const float* fn = nsum + (size_t)row  * D_C;
    const float* ws = Wself  + (size_t)col * D_C;
    const float* wn = Wneigh + (size_t)col * D_C;

    v8f c = {};
#pragma unroll
    for (int k0 = 0; k0 < D_C; k0 += 4) {
        int kk = k0 + 2 * half;
        v2f a_s = *(const v2f*)(fs + kk);
        v2f b_s = *(const v2f*)(ws + kk);
        v2f a_nv = *(const v2f*)(fn + kk);
        v2f b_n = *(const v2f*)(wn + kk);
        v2f a_n;
        a_n.x = a_nv.x * rdeg;
        a_n.y = a_nv.y * rdeg;
        c = __builtin_amdgcn_wmma_f32_16x16x4_f32(
                false, a_s, false, b_s, (short)0, c, false, false);
        c = __builtin_amdgcn_wmma_f32_16x16x4_f32(
                false, a_n, false, b_n, (short)0, c, false, false);
    }

    float bias = bself[col] + bneigh[col];
    // C/D layout: VGPR r -> M = tm*16 + r + 8*half, N = col
#pragma unroll
    for (int r = 0; r < 8; ++r) {
        int m = tm * 16 + r + 8 * half;
        out[(size_t)m * D_C + col] = c[r] + bias;
    }
}

extern "C" void kernel_launch(void* const* d_in, const int* in_sizes, int n_in,
                              void* d_out, int out_size, void* d_ws, size_t ws_size,
                              hipStream_t stream) {
    const float* feat     = (const float*)d_in[0];
    const int*   esrc     = (const int*)  d_in[1];
    const int*   edst     = (const int*)  d_in[2];
    const int*   self_src = (const int*)  d_in[3];
    const float* Wself    = (const float*)d_in[4];
    const float* bself    = (const float*)d_in[5];
    const float* Wneigh   = (const float*)d_in[6];
    const float* bneigh   = (const float*)d_in[7];
    float*       out      = (float*)d_out;
    const int E = in_sizes[1];

    float* nsum = (float*)d_ws;                       // [N_DST, 64]
    float* deg  = nsum + (size_t)N_DST_C * D_C;       // [N_DST]

    // 1) zero accumulators (3,250,000 floats, divisible by 4)
    int ztot = N_DST_C * D_C + N_DST_C;
    int n4 = ztot / 4;
    sage_zero_ws<<<(n4 + 255) / 256, 256, 0, stream>>>((float4*)d_ws, n4);

    // 2) edge scatter: E*64 threads
    long long sthreads = (long long)E * D_C;
    int sblocks = (int)((sthreads + 255) / 256);
    sage_scatter<<<sblocks, 256, 0, stream>>>(feat, esrc, edst, nsum, deg, E);

    // 3) fused dual GEMM via f32 WMMA: one wave per 16x16 tile
    int waves = (N_DST_C / 16) * (D_C / 16);          // 12500
    int gblocks = (waves * 32 + 255) / 256;           // 8 waves / block
    sage_gemm_wmma<<<gblocks, 256, 0, stream>>>(feat, self_src, nsum, deg,
                                                Wself, bself, Wneigh, bneigh, out);
}